// ACT_R_34943853920367
// MI455X (gfx1250) — compile-verified
//
#include <hip/hip_runtime.h>

// ACT-R activation recurrence for MI455X (gfx1250, wave32).
//
// Reformulation: S_i = sum_{j<i} max((sp_i-sp_j)*H,1)^(-(w1*S_j+w0)), S_0 = 0,
// out_{i-1} = p/(p+C), p = S_i^{1/S}, C = e^{TAU/S}.   (exp(m_j) == S_j, so the
// recurrence itself needs no transcendentals; only the pow terms do.)
//
// One block per batch column. i tiled by 16; bulk (j < i0) terms are computed
// 16x4 at a time per wave and row-summed with V_WMMA_F32_16X16X4_F32
// (A = terms, B = ones). In-tile 16x16 triangle fixed up serially by wave 0.
// sp column staged to LDS via gfx1250 async global->LDS loads when available.

typedef __attribute__((ext_vector_type(2))) float v2f;
typedef __attribute__((ext_vector_type(8))) float v8f;

#define L_CONST  1024
#define TI       16
#define NTHREADS 512
#define NWAVES   (NTHREADS / 32)

__device__ __forceinline__ float fast_log2(float x) { return __builtin_amdgcn_logf(x); }
__device__ __forceinline__ float fast_exp2(float x) { return __builtin_amdgcn_exp2f(x); }

#if __has_builtin(__builtin_amdgcn_global_load_async_to_lds_b32)
#define HAVE_ASYNC_LDS 1
typedef __attribute__((address_space(3))) int lds_int;
typedef __attribute__((address_space(1))) int glb_int;
__device__ __forceinline__ lds_int* to_lds(void* p) {
  return (lds_int*)(unsigned int)(unsigned long long)p;   // generic LDS ptr low 32b = LDS offset
}
__device__ __forceinline__ glb_int* to_glb(const void* p) {
  return (glb_int*)(unsigned long long)p;
}
#else
#define HAVE_ASYNC_LDS 0
#endif

__global__ __launch_bounds__(NTHREADS)
void actr_wmma_kernel(const float* __restrict__ sp,
                      const float* __restrict__ w,
                      float* __restrict__ out,
                      int B) {
  __shared__ float sp_s[L_CONST];          // sp column * H for this batch
  __shared__ float nd_s[L_CONST];          // negdecay[j] = -(w1*S_j + w0)
  __shared__ float wsum[NWAVES][TI];       // per-wave tile row-sums

  const int b    = blockIdx.x;
  const int tid  = threadIdx.x;
  const int lane = tid & 31;
  const int wave = __builtin_amdgcn_readfirstlane(tid >> 5);  // scalar wave id

  const float w0   = w[0];
  const float w1   = w[1];
  const float Hc   = 2160.0f;                       // 86400 * 0.025
  const float TAUc = -0.704205679427144f;
  const float Sc   = 0.254893976981164f;
  const float invS = 1.0f / Sc;
  const float Ck   = fast_exp2(TAUc * invS * 1.4426950408889634f); // e^{TAU/S}

  // ---- stage this batch's sp column into LDS ----
#if HAVE_ASYNC_LDS
  for (int l = tid; l < L_CONST; l += NTHREADS)
    __builtin_amdgcn_global_load_async_to_lds_b32(to_glb(sp + (size_t)l * B + b),
                                                  to_lds(&sp_s[l]), 0, 0);
#if __has_builtin(__builtin_amdgcn_s_wait_asynccnt)
  __builtin_amdgcn_s_wait_asynccnt(0);
#else
  asm volatile("s_wait_asynccnt 0" ::: "memory");
#endif
  __syncthreads();
  for (int l = tid; l < L_CONST; l += NTHREADS)    // pre-scale by H in place
    sp_s[l] = sp_s[l] * Hc;
#else
  for (int l = tid; l < L_CONST; l += NTHREADS)
    sp_s[l] = sp[(size_t)l * B + b] * Hc;
#endif
  __syncthreads();

  // WMMA A-matrix (16x4 f32) lane mapping: M = lane&15;
  // lanes 0-15 hold K=0 (vgpr0) / K=1 (vgpr1); lanes 16-31 hold K=2 / K=3.
  const int mi   = lane & 15;
  const int koff = (lane >> 4) << 1;   // 0 or 2

  v2f bones;
  bones.x = 1.0f;
  bones.y = 1.0f;

  for (int i0 = 0; i0 < L_CONST; i0 += TI) {
    const float spi = sp_s[i0 + mi];     // (already *H) the i-row this lane evaluates

    // ---- bulk phase: j in [0, i0), chunks of 4 j's per wave-step ----
    v8f acc = {0.f, 0.f, 0.f, 0.f, 0.f, 0.f, 0.f, 0.f};
#pragma unroll 2
    for (int j0 = 4 * wave; j0 < i0; j0 += 4 * NWAVES) {
      // i0 is a multiple of 16 and chunks are 4-aligned -> never crosses i0.
      const int ja = j0 + koff;
      const int jb = ja + 1;
      const float da = fmaxf(spi - sp_s[ja], 1.0f);
      const float db = fmaxf(spi - sp_s[jb], 1.0f);
      v2f a;
      a.x = fast_exp2(nd_s[ja] * fast_log2(da));
      a.y = fast_exp2(nd_s[jb] * fast_log2(db));
      // D[m][n] += sum_k A[m][k]  (B = ones): row-sums accumulate in every column.
      acc = __builtin_amdgcn_wmma_f32_16x16x4_f32(
          false, a, false, bones, (short)0, acc, false, false);
    }

    // Column 0 of D: lane 0 holds M=0..7 in vgprs 0..7, lane 16 holds M=8..15.
    if (lane == 0) {
#pragma unroll
      for (int r = 0; r < 8; ++r) wsum[wave][r] = acc[r];
    } else if (lane == 16) {
#pragma unroll
      for (int r = 0; r < 8; ++r) wsum[wave][8 + r] = acc[r];
    }
    __syncthreads();

    // ---- triangle phase: wave 0, lanes 0..15 each own one i of the tile ----
    if (wave == 0 && lane < 16) {
      const int i = i0 + lane;
      float s = 0.0f;
#pragma unroll
      for (int wv = 0; wv < NWAVES; ++wv) s += wsum[wv][lane];

      const float spl = sp_s[i];
#pragma unroll
      for (int t = 0; t < TI - 1; ++t) {
        // S_j for j = i0+t is finalized in lane t at this point (uniform t -> v_readlane).
        const float Sj = __int_as_float(__builtin_amdgcn_readlane(__float_as_int(s), t));
        const float nd = -(w1 * Sj + w0);
        if (lane > t) {
          const float d = fmaxf(spl - sp_s[i0 + t], 1.0f);
          s += fast_exp2(nd * fast_log2(d));
        }
      }
      nd_s[i] = -(w1 * s + w0);          // publish for later tiles
      if (i > 0) {
        const float p = fast_exp2(fast_log2(s) * invS);   // S_i^{1/S}
        out[(size_t)(i - 1) * B + b] = p / (p + Ck);
      }
    }
    __syncthreads();
  }
}

extern "C" void kernel_launch(void* const* d_in, const int* in_sizes, int n_in,
                              void* d_out, int out_size, void* d_ws, size_t ws_size,
                              hipStream_t stream) {
  const float* sp = (const float*)d_in[0];   // [L, B, 1] fp32
  const float* w  = (const float*)d_in[1];   // [2] fp32
  float* out = (float*)d_out;                // [L-1, B, 1] fp32
  const int B = in_sizes[0] / L_CONST;       // 256
  actr_wmma_kernel<<<dim3(B), dim3(NTHREADS), 0, stream>>>(sp, w, out, B);
}